// Partitioner_85839216377874
// MI455X (gfx1250) — compile-verified
//
#include <hip/hip_runtime.h>
#include <hip/hip_bf16.h>
#include <math.h>

// ---------------- problem constants (from reference) ----------------
#define BB        8
#define NN        2048
#define DD        128
#define PROJ      64
#define THETA_IN  387      // D + NHID + 3
#define THETA_HID 128
#define SEED_CTX  384      // D + NHID
#define ALPHA_C   0.2f
#define TAU_C     1.0f
#define KCAP      2        // CLUSTER_SIZE_MAX - 1

typedef __attribute__((ext_vector_type(2))) float v2f;
typedef __attribute__((ext_vector_type(8))) float v8f;

// =====================================================================
// Kernel 1: per-batch seed argmax, q = Wq @ x[seed], theta MLP head
// grid = B, block = 128
// =====================================================================
__global__ __launch_bounds__(128) void k_seed_q_theta(
    const float* __restrict__ x,            // [B,N,D]
    const float* __restrict__ seed_ctx,     // [B,384]
    const float* __restrict__ local_stats,  // [B,3]
    const float* __restrict__ cluster_mask, // [B,N]
    const float* __restrict__ Wq,           // [64,128]
    const float* __restrict__ W1,           // [387,128]
    const float* __restrict__ b1,           // [128]
    const float* __restrict__ W2,           // [128,1]
    const float* __restrict__ b2,           // [1]
    float* __restrict__ q_ws,               // [B,64]
    float* __restrict__ theta_ws,           // [B]
    float* __restrict__ theta_out)          // [B] (slice of d_out)
{
    const int b   = blockIdx.x;
    const int tid = threadIdx.x;

    __shared__ float sv[128];
    __shared__ int   si[128];
    __shared__ float tin[THETA_IN];
    __shared__ float sh[128];
    __shared__ int   s_seed;

    // ---- argmax(cluster_mask[b]) with first-occurrence tie-break ----
    float bv = -1e30f; int bi = 0x7fffffff;
    for (int i = tid; i < NN; i += 128) {
        float v = cluster_mask[b * NN + i];
        if (v > bv || (v == bv && i < bi)) { bv = v; bi = i; }
    }
    sv[tid] = bv; si[tid] = bi;
    __syncthreads();
    for (int s = 64; s > 0; s >>= 1) {
        if (tid < s) {
            if (sv[tid + s] > sv[tid] ||
                (sv[tid + s] == sv[tid] && si[tid + s] < si[tid])) {
                sv[tid] = sv[tid + s]; si[tid] = si[tid + s];
            }
        }
        __syncthreads();
    }
    if (tid == 0) s_seed = si[0];
    __syncthreads();
    const int seed = s_seed;

    // ---- q[b,h] = sum_d Wq[h,d] * x[b,seed,d] ----
    const float* sx = x + ((size_t)b * NN + seed) * DD;
    if (tid < PROJ) {
        float acc = 0.0f;
        #pragma unroll 4
        for (int d = 0; d < DD; ++d) acc += Wq[tid * DD + d] * sx[d];
        q_ws[b * PROJ + tid] = acc;
    }

    // ---- theta head: relu(tin @ W1 + b1) @ W2 + b2 ----
    for (int i = tid; i < SEED_CTX; i += 128) tin[i] = seed_ctx[b * SEED_CTX + i];
    if (tid < 3) tin[SEED_CTX + tid] = local_stats[b * 3 + tid];
    __syncthreads();

    float h = b1[tid];
    for (int i = 0; i < THETA_IN; ++i) h += tin[i] * W1[i * THETA_HID + tid];
    h = fmaxf(h, 0.0f);
    sh[tid] = h * W2[tid];
    __syncthreads();
    for (int s = 64; s > 0; s >>= 1) {
        if (tid < s) sh[tid] += sh[tid + s];
        __syncthreads();
    }
    if (tid == 0) {
        float th = sh[0] + b2[0];
        theta_ws[b]  = th;
        theta_out[b] = th;
    }
}

// =====================================================================
// Kernel 2: WMMA x-projection scores + adj streaming (e2c/deg) + sigmoid
// grid = (N/128, B), block = 256 (8 waves; each wave owns 16 nodes)
// =====================================================================
__global__ __launch_bounds__(256) void k_scores_gate(
    const float* __restrict__ x,            // [B,N,D]
    const float* __restrict__ adj,          // [B,N,N]
    const float* __restrict__ cluster_mask, // [B,N]
    const float* __restrict__ cand_mask,    // [B,N]
    const float* __restrict__ Wk,           // [64,128]
    const float* __restrict__ q_ws,         // [B,64]
    const float* __restrict__ theta_ws,     // [B]
    float* __restrict__ p_out)              // [B,N] (p slice of d_out)
{
    __shared__ float sWk[PROJ * DD];        // 32 KB
    __shared__ float sMask[NN];             // 8 KB
    __shared__ float sScore[128];

    const int b     = blockIdx.y;
    const int tile0 = blockIdx.x * 128;
    const int tid   = threadIdx.x;
    const int wave  = tid >> 5;
    const int lane  = tid & 31;
    const int half  = lane >> 4;            // K-pair select (0: K=0,1 | 1: K=2,3)
    const int r     = lane & 15;            // M (for A) / N (for B) index

    for (int i = tid; i < PROJ * DD; i += 256) sWk[i] = Wk[i];
    for (int i = tid; i < NN; i += 256)        sMask[i] = cluster_mask[b * NN + i];
    __syncthreads();

    // ---------------- phase A: x_k tile via V_WMMA_F32_16X16X4_F32 ----
    const int    n0 = tile0 + wave * 16;
    const float* xb = x + ((size_t)b * NN + n0) * DD;

    v8f acc0 = {}, acc1 = {}, acc2 = {}, acc3 = {};
    for (int k0 = 0; k0 < DD; k0 += 4) {
        const int kk = k0 + 2 * half;
        v2f a;                                    // A: 16x4 f32 fragment
        a.x = xb[r * DD + kk];
        a.y = xb[r * DD + kk + 1];
        v2f f0, f1, f2, f3;                       // B: 4x16 f32 fragments (Wk^T)
        f0.x = sWk[( 0 + r) * DD + kk]; f0.y = sWk[( 0 + r) * DD + kk + 1];
        f1.x = sWk[(16 + r) * DD + kk]; f1.y = sWk[(16 + r) * DD + kk + 1];
        f2.x = sWk[(32 + r) * DD + kk]; f2.y = sWk[(32 + r) * DD + kk + 1];
        f3.x = sWk[(48 + r) * DD + kk]; f3.y = sWk[(48 + r) * DD + kk + 1];
        acc0 = __builtin_amdgcn_wmma_f32_16x16x4_f32(false, a, false, f0, (short)0, acc0, false, false);
        acc1 = __builtin_amdgcn_wmma_f32_16x16x4_f32(false, a, false, f1, (short)0, acc1, false, false);
        acc2 = __builtin_amdgcn_wmma_f32_16x16x4_f32(false, a, false, f2, (short)0, acc2, false, false);
        acc3 = __builtin_amdgcn_wmma_f32_16x16x4_f32(false, a, false, f3, (short)0, acc3, false, false);
    }

    // scores[n] = (x_k[n,:] . q) / sqrt(64); lane's N-column = ht*16 + r
    const float q0 = q_ws[b * PROJ +  0 + r];
    const float q1 = q_ws[b * PROJ + 16 + r];
    const float q2 = q_ws[b * PROJ + 32 + r];
    const float q3 = q_ws[b * PROJ + 48 + r];

    float part[8];
    #pragma unroll
    for (int rr = 0; rr < 8; ++rr)
        part[rr] = acc0[rr] * q0 + acc1[rr] * q1 + acc2[rr] * q2 + acc3[rr] * q3;
    #pragma unroll
    for (int m = 1; m < 16; m <<= 1) {
        #pragma unroll
        for (int rr = 0; rr < 8; ++rr)
            part[rr] += __shfl_xor(part[rr], m, 16);
    }
    if (r == 0) {
        #pragma unroll
        for (int rr = 0; rr < 8; ++rr)
            sScore[wave * 16 + half * 8 + rr] = part[rr] * 0.125f;   // 1/sqrt(PROJ)
    }
    __syncthreads();

    // ---------------- phase B: stream adj rows (the 128 MB term) -----
    const float theta   = theta_ws[b];
    const float inv_tau = 1.0f / fmaxf(TAU_C, 1e-6f);

    for (int j = 0; j < 16; ++j) {
        const int    n   = n0 + j;
        const float* row = adj + ((size_t)b * NN + n) * NN;
        float s_all = 0.0f, s_msk = 0.0f;
        for (int c = lane * 4; c < NN; c += 128) {       // 32 lanes * float4
            const float4 v = *(const float4*)(row + c);
            const float4 m = *(const float4*)(sMask + c);
            s_all += v.x + v.y + v.z + v.w;
            s_msk += v.x * m.x + v.y * m.y + v.z * m.z + v.w * m.w;
        }
        #pragma unroll
        for (int m = 1; m < 32; m <<= 1) {
            s_all += __shfl_xor(s_all, m, 32);
            s_msk += __shfl_xor(s_msk, m, 32);
        }
        if (lane == 0) {
            const float deg   = fmaxf(s_all, 1.0f);
            const float cand  = sScore[wave * 16 + j] + ALPHA_C * (s_msk / deg);
            const float logit = (cand - theta) * inv_tau;
            const float pp    = (1.0f / (1.0f + expf(-logit))) * cand_mask[b * NN + n];
            p_out[b * NN + n] = pp;
        }
    }
}

// =====================================================================
// Kernel 3: per-batch count(p>0.5), top-2 (stable), trim, hard gate
// grid = B, block = 256; reads/writes the p slice in-place, writes hard
// =====================================================================
__global__ __launch_bounds__(256) void k_trim_hard(float* __restrict__ out)
{
    const int b   = blockIdx.x;
    const int tid = threadIdx.x;
    float* p    = out + (size_t)BB * NN + (size_t)b * NN;  // p slice
    float* hard = out + (size_t)b * NN;                    // hard slice

    __shared__ float svv[256];
    __shared__ int   sii[256];
    __shared__ int   scc[256];
    __shared__ int   s_idx1, s_idx2, s_cnt;

    // pass 1: count + argmax (first occurrence)
    float bv = -1.0f; int bi = NN; int cnt = 0;
    for (int i = tid; i < NN; i += 256) {
        const float v = p[i];
        cnt += (v > 0.5f) ? 1 : 0;
        if (v > bv || (v == bv && i < bi)) { bv = v; bi = i; }
    }
    svv[tid] = bv; sii[tid] = bi; scc[tid] = cnt;
    __syncthreads();
    for (int s = 128; s > 0; s >>= 1) {
        if (tid < s) {
            scc[tid] += scc[tid + s];
            if (svv[tid + s] > svv[tid] ||
                (svv[tid + s] == svv[tid] && sii[tid + s] < sii[tid])) {
                svv[tid] = svv[tid + s]; sii[tid] = sii[tid + s];
            }
        }
        __syncthreads();
    }
    if (tid == 0) { s_idx1 = sii[0]; s_cnt = scc[0]; }
    __syncthreads();
    const int idx1 = s_idx1;

    // pass 2: second-best excluding idx1
    bv = -1.0f; bi = NN;
    for (int i = tid; i < NN; i += 256) {
        if (i == idx1) continue;
        const float v = p[i];
        if (v > bv || (v == bv && i < bi)) { bv = v; bi = i; }
    }
    svv[tid] = bv; sii[tid] = bi;
    __syncthreads();
    for (int s = 128; s > 0; s >>= 1) {
        if (tid < s) {
            if (svv[tid + s] > svv[tid] ||
                (svv[tid + s] == svv[tid] && sii[tid + s] < sii[tid])) {
                svv[tid] = svv[tid + s]; sii[tid] = sii[tid + s];
            }
        }
        __syncthreads();
    }
    if (tid == 0) s_idx2 = sii[0];
    __syncthreads();
    const int  idx2      = s_idx2;
    const bool need_trim = (s_cnt - KCAP) > 0;

    for (int i = tid; i < NN; i += 256) {
        float v = p[i];
        if (need_trim && i != idx1 && i != idx2) v = 0.0f;
        p[i]    = v;
        hard[i] = (v > 0.5f) ? 1.0f : 0.0f;   // + (p - stop_grad(p)) == 0
    }
}

// =====================================================================
extern "C" void kernel_launch(void* const* d_in, const int* in_sizes, int n_in,
                              void* d_out, int out_size, void* d_ws, size_t ws_size,
                              hipStream_t stream) {
    const float* x            = (const float*)d_in[0];   // [B,N,D]
    const float* adj          = (const float*)d_in[1];   // [B,N,N]
    const float* seed_ctx     = (const float*)d_in[2];   // [B,384]
    const float* local_stats  = (const float*)d_in[3];   // [B,3]
    const float* cluster_mask = (const float*)d_in[4];   // [B,N]
    const float* cand_mask    = (const float*)d_in[5];   // [B,N]
    const float* Wq           = (const float*)d_in[6];   // [64,128]
    const float* Wk           = (const float*)d_in[7];   // [64,128]
    const float* W1           = (const float*)d_in[8];   // [387,128]
    const float* b1           = (const float*)d_in[9];   // [128]
    const float* W2           = (const float*)d_in[10];  // [128,1]
    const float* b2           = (const float*)d_in[11];  // [1]

    float* out      = (float*)d_out;                     // hard | p | theta
    float* q_ws     = (float*)d_ws;                      // [B,64]
    float* theta_ws = q_ws + BB * PROJ;                  // [B]
    float* theta_out = out + (size_t)2 * BB * NN;

    k_seed_q_theta<<<BB, 128, 0, stream>>>(
        x, seed_ctx, local_stats, cluster_mask, Wq, W1, b1, W2, b2,
        q_ws, theta_ws, theta_out);

    k_scores_gate<<<dim3(NN / 128, BB), 256, 0, stream>>>(
        x, adj, cluster_mask, cand_mask, Wk, q_ws, theta_ws,
        out + (size_t)BB * NN);

    k_trim_hard<<<BB, 256, 0, stream>>>(out);
}